// Roe_8787503087637
// MI455X (gfx1250) — compile-verified
//
#include <hip/hip_runtime.h>
#include <cstdint>

// Roe flux 1-D stencil, (m=4096, 3, l=4096) fp32, for MI455X (gfx1250).
// - Tile of one row async-staged into LDS (GLOBAL_LOAD_ASYNC_TO_LDS_B128/B32,
//   ASYNCcnt) -> no VGPR round-trip for the stencil halo.
// - Per-interface factored Roe solve: the Roe average is shared by the two
//   adjacent cells' flux_side() calls, halving sqrt/rcp count.
// - v_rcp_f32 + 1 Newton step and raw v_sqrt_f32 replace the ~10-op IEEE
//   division sequences (kernel is VALU/transcendental-bound: ~7.5 GFLOP of
//   fp32 vs a 1.2 GB / ~52us HBM floor at 23.3 TB/s).

#define TILE     1024
#define NTHREADS 256
#define SOFF     4                 // first interior cell at physical index 4 (16B aligned)
#define NPHYS    (SOFF + TILE + 4) // halos live at [SOFF-1] and [SOFF+TILE]
#define NIFACE   (TILE + 1)

__device__ __forceinline__ void async_ld_b128(uint32_t lds_byte_addr, const float* g) {
    asm volatile("global_load_async_to_lds_b128 %0, %1, off"
                 :: "v"(lds_byte_addr), "v"(g) : "memory");
}
__device__ __forceinline__ void async_ld_b32(uint32_t lds_byte_addr, const float* g) {
    asm volatile("global_load_async_to_lds_b32 %0, %1, off"
                 :: "v"(lds_byte_addr), "v"(g) : "memory");
}
__device__ __forceinline__ uint32_t lds_addr_of(const void* p) {
    return (uint32_t)(uintptr_t)p;   // generic shared ptr: low 32 bits = LDS byte addr
}
__device__ __forceinline__ float fast_rcp(float x) {
    float r = __builtin_amdgcn_rcpf(x);      // v_rcp_f32 (~1 ulp)
    return r * (2.0f - x * r);               // 1 Newton step -> ~fp32 accurate
}
__device__ __forceinline__ float fast_sqrt(float x) {
    return __builtin_amdgcn_sqrtf(x);        // v_sqrt_f32
}

__global__ __launch_bounds__(NTHREADS)
void roe_flux_kernel(const float* __restrict__ u, float* __restrict__ out,
                     int l) {
    __shared__ __align__(16) float s_rho[NPHYS], s_mom[NPHYS], s_E[NPHYS];
    __shared__ __align__(16) float s_s[NPHYS], s_mos[NPHYS], s_sH[NPHYS];
    __shared__ __align__(16) float s_Fm[3][NIFACE + 3], s_Fp[3][NIFACE + 3];

    const int row = blockIdx.y;
    const int x0  = blockIdx.x * TILE;       // requires l % TILE == 0 (l = 4096)
    const int tid = threadIdx.x;

    const float* __restrict__ rowb = u + (size_t)row * 3u * (size_t)l;

    // ---- Phase 0: async-stage tile (+clamped halo) into LDS ----------------
    {
        // Interior: 256 lanes x float4 = 1024 floats per component, all 16B aligned.
        const int c4 = tid * 4;
        async_ld_b128(lds_addr_of(&s_rho[SOFF + c4]), rowb +            x0 + c4);
        async_ld_b128(lds_addr_of(&s_mom[SOFF + c4]), rowb + l        + x0 + c4);
        async_ld_b128(lds_addr_of(&s_E  [SOFF + c4]), rowb + 2 * l    + x0 + c4);
        // Halos: lane 0 -> left (x0-1 clamped), lane 1 -> right (x0+TILE clamped).
        if (tid < 2) {
            int x  = (tid == 0) ? (x0 - 1) : (x0 + TILE);
            x      = (x < 0) ? 0 : ((x >= l) ? (l - 1) : x);
            int ph = (tid == 0) ? (SOFF - 1) : (SOFF + TILE);
            async_ld_b32(lds_addr_of(&s_rho[ph]), rowb +         x);
            async_ld_b32(lds_addr_of(&s_mom[ph]), rowb + l     + x);
            async_ld_b32(lds_addr_of(&s_E  [ph]), rowb + 2 * l + x);
        }
    }
    asm volatile("s_wait_asynccnt 0" ::: "memory");
    __syncthreads();

    // ---- Phase 1: per-point primitives: s=sqrt(rho), mo/s, s*H -------------
    for (int p = SOFF - 1 + tid; p <= SOFF + TILE; p += NTHREADS) {
        float rho = s_rho[p], mo = s_mom[p], E = s_E[p];
        float inv_rho = fast_rcp(rho);
        float pr = 0.4f * (E - 0.5f * mo * mo * inv_rho);   // (gamma-1)=0.4
        float H  = (E + pr) * inv_rho;
        float s  = fast_sqrt(rho);
        s_s[p]   = s;
        s_mos[p] = mo * s * inv_rho;   // mo / sqrt(rho)
        s_sH[p]  = s * H;
    }
    __syncthreads();

    // ---- Phase 2: per-interface Roe flux, both upwind signs ----------------
    // Interface j (0..TILE) couples physical cells 3+j and 4+j. One Roe solve
    // yields both (lambda - |lambda|) and (lambda + |lambda|) flux halves.
    for (int j = tid; j < NIFACE; j += NTHREADS) {
        const int a = SOFF - 1 + j, b = SOFF + j;
        float invS = fast_rcp(s_s[a] + s_s[b]);
        float ua = (s_mos[a] + s_mos[b]) * invS;
        float Ha = (s_sH[a]  + s_sH[b])  * invS;
        float c2  = 0.5f * ua * ua;
        float ca2 = 0.4f * (Ha - c2);
        float ca  = fast_sqrt(ca2);
        float c1  = 0.2f * fast_rcp(ca2);    // 0.5*(gamma-1)/ca^2
        float c3  = ca * 2.5f;               // ca/(gamma-1)

        float d0 = s_rho[b] - s_rho[a];
        float d1 = s_mom[b] - s_mom[a];
        float d2 = s_E[b]   - s_E[a];

        float l1 = ua - ca, l3 = ua + ca;
        float P0 = c1 * ((c2 + ua * c3) * d0 - (ua + c3) * d1 + d2);
        float P1 = 2.0f * c1 * ((ca * c3 - c2) * d0 + ua * d1 - d2);
        float P2 = c1 * ((c2 - ua * c3) * d0 - (ua - c3) * d1 + d2);

        float a1 = fabsf(l1), a2 = fabsf(ua), a3 = fabsf(l3);
        float m0 = (l1 - a1) * P0, m1 = (ua - a2) * P1, m2 = (l3 - a3) * P2;
        float p0 = (l1 + a1) * P0, p1 = (ua + a2) * P1, p2 = (l3 + a3) * P2;

        float Hlo = Ha - ua * ca, Hhi = Ha + ua * ca;
        s_Fm[0][j] = m0 + m1 + m2;                      // used by left cell (sign=-1)
        s_Fm[1][j] = l1 * m0 + ua * m1 + l3 * m2;
        s_Fm[2][j] = Hlo * m0 + c2 * m1 + Hhi * m2;
        s_Fp[0][j] = p0 + p1 + p2;                      // used by right cell (sign=+1)
        s_Fp[1][j] = l1 * p0 + ua * p1 + l3 * p2;
        s_Fp[2][j] = Hlo * p0 + c2 * p1 + Hhi * p2;
    }
    __syncthreads();

    // ---- Phase 3: combine per cell, vectorized float4 stores ---------------
    // out[c] = -(Fm[c+1] + Fp[c]) / (2*DX), DX=0.005 -> factor -100.
    {
        float* __restrict__ orow = out + (size_t)row * 3u * (size_t)l;
        const float kinv = -100.0f;
        const int c = tid * 4;
        #pragma unroll
        for (int k = 0; k < 3; ++k) {
            float4 o;
            o.x = kinv * (s_Fm[k][c + 1] + s_Fp[k][c]);
            o.y = kinv * (s_Fm[k][c + 2] + s_Fp[k][c + 1]);
            o.z = kinv * (s_Fm[k][c + 3] + s_Fp[k][c + 2]);
            o.w = kinv * (s_Fm[k][c + 4] + s_Fp[k][c + 3]);
            *reinterpret_cast<float4*>(orow + (size_t)k * l + x0 + c) = o;
        }
    }
}

extern "C" void kernel_launch(void* const* d_in, const int* in_sizes, int n_in,
                              void* d_out, int out_size, void* d_ws, size_t ws_size,
                              hipStream_t stream) {
    const float* u = (const float*)d_in[0];
    float* out = (float*)d_out;
    const int l = 4096;                       // reference shape (m, 3, l)
    const int m = in_sizes[0] / (3 * l);
    dim3 grid(l / TILE, m);
    roe_flux_kernel<<<grid, NTHREADS, 0, stream>>>(u, out, l);
}